// MixModel_37692632989960
// MI455X (gfx1250) — compile-verified
//
#include <hip/hip_runtime.h>

typedef float v2f __attribute__((ext_vector_type(2)));
typedef float v8f __attribute__((ext_vector_type(8)));

#define DIM 36
#define J18 18
#define ROWS_PER_WAVE 16
#define WAVES 4
#define TILE_F (ROWS_PER_WAVE * DIM) /* 576 floats per wave-tile */
#define TILE_V4 (TILE_F / 4)         /* 144 float4 per wave-tile */

// D = A(16x4 f32) * B(4x16 f32) + C(16x16 f32), full fp32 matrix op.
__device__ __forceinline__ v8f wmma4(v2f a, v2f b, v8f c) {
  return __builtin_amdgcn_wmma_f32_16x16x4_f32(false, a, false, b, (short)0, c,
                                               false, false);
}

// Single-instruction ReLU (inputs are never sNaN, skip canonicalize).
__device__ __forceinline__ float relu1(float x) {
  float y;
  asm("v_max_num_f32 %0, 0, %1" : "=v"(y) : "v"(x));
  return y;
}

// ReLU only the first NREG accumulator registers (the ones consumed by the
// next layer's K-chunks); the rest are dead.
template <int NREG>
__device__ __forceinline__ v8f reluN(v8f x) {
#pragma unroll
  for (int r = 0; r < NREG; ++r) x[r] = relu1(x[r]);
  return x;
}

// B operand chunk c taken DIRECTLY from the previous accumulator register
// pair (d[2c], d[2c+1]) -- no cross-lane data movement.
template <int C>
__device__ __forceinline__ v2f bpair(v8f d) {
  v2f b;
  b.x = d[2 * C];
  b.y = d[2 * C + 1];
  return b;
}

// Row permutation: producer stores feature f at D-row q(f), with
// q = [0,1,8,9,2,3,10,11,4,5,12,13,6,7,14,15].  qinv(p) is its inverse.
// With this q, the consumer's K-index order over raw accumulator pairs is
// exactly the natural dense order (qinv(2c+v+8h) == 4c+v+2h == k), so all
// weight loads stay dense in K and chunk count is ceil(I/4).
__device__ __forceinline__ int qinv(int p) {
  return 4 * ((p >> 1) & 3) + (p & 1) + 2 * (p >> 3);
}

// Dense A operand, natural output rows (final layers): chunk c of W[O x I].
__device__ __forceinline__ v2f load_a(const float* W, int O, int I, int c,
                                      int lane) {
  int m = lane & 15;
  int kb = 4 * c + ((lane & 16) ? 2 : 0);
  v2f a;
  a.x = (m < O && kb < I) ? W[m * I + kb] : 0.0f;
  a.y = (m < O && (kb + 1) < I) ? W[m * I + kb + 1] : 0.0f;
  return a;
}

// Dense-K A operand with q-permuted output rows (hidden layers).
__device__ __forceinline__ v2f load_a_q(const float* W, int O, int I, int c,
                                        int lane) {
  int m = qinv(lane & 15);
  int kb = 4 * c + ((lane & 16) ? 2 : 0);
  v2f a;
  a.x = (m < O && kb < I) ? W[m * I + kb] : 0.0f;
  a.y = (m < O && (kb + 1) < I) ? W[m * I + kb + 1] : 0.0f;
  return a;
}

// Bias as C operand, natural rows (final layers).
__device__ __forceinline__ v8f load_bias(const float* b, int O, int lane) {
  int mb = (lane & 16) ? 8 : 0;
  v8f c;
#pragma unroll
  for (int r = 0; r < 8; ++r) c[r] = ((r + mb) < O) ? b[r + mb] : 0.0f;
  return c;
}

// Bias as C operand with q-permuted rows (hidden layers).
__device__ __forceinline__ v8f load_bias_q(const float* b, int O, int lane) {
  int mb = (lane & 16) ? 8 : 0;
  v8f c;
#pragma unroll
  for (int r = 0; r < 8; ++r) {
    int f = qinv(r + mb);
    c[r] = (f < O) ? b[f] : 0.0f;
  }
  return c;
}

__global__ __launch_bounds__(WAVES * 32) void mixmodel_kernel(
    const float* __restrict__ u, float* __restrict__ out,
    const float* __restrict__ reg1, const float* __restrict__ reg2,
    const float* __restrict__ W1, const float* __restrict__ b1,
    const float* __restrict__ W2, const float* __restrict__ b2,
    const float* __restrict__ W3, const float* __restrict__ b3,
    const float* __restrict__ W4, const float* __restrict__ b4,
    const float* __restrict__ W5, const float* __restrict__ b5,
    const float* __restrict__ V1, const float* __restrict__ c1,
    const float* __restrict__ V2, const float* __restrict__ c2,
    const float* __restrict__ V3, const float* __restrict__ c3, int N) {
  __shared__ __align__(16) float lds[WAVES * TILE_F * 2];
  const int tid = threadIdx.x;
  const int wave = tid >> 5;
  const int lane = tid & 31;
  const int inBase = wave * (TILE_F * 2);
  const int outBase = inBase + TILE_F;
  const long rowBase = ((long)blockIdx.x * WAVES + wave) * ROWS_PER_WAVE;
  const long elemBase = rowBase * DIM;
  long availL = (long)N * DIM - elemBase;
  const int avail = (availL > TILE_F) ? TILE_F : (availL < 0 ? 0 : (int)availL);
  const int avail4 = avail >> 2;  // avail is a multiple of 36 -> of 4

  // --- cooperative, coalesced float4 load of this wave's 16x36 input tile ---
  const float4* src4 = reinterpret_cast<const float4*>(u + elemBase);
  float4* lin4 = reinterpret_cast<float4*>(&lds[inBase]);
#pragma unroll
  for (int i = lane; i < TILE_V4; i += 32) {
    float4 v = make_float4(0.0f, 0.0f, 0.0f, 0.0f);
    if (i < avail4) v = src4[i];
    lin4[i] = v;
  }
  __syncthreads();

  // --- persistent weight / bias operand registers ---
  // Hidden layers: q-permuted output rows, dense K. Final layers: natural.
  v2f aW1 = load_a_q(W1, 5, 3, 0, lane);
  v2f aW2_0 = load_a_q(W2, 10, 5, 0, lane),
      aW2_1 = load_a_q(W2, 10, 5, 1, lane);
  v2f aW3_0 = load_a_q(W3, 15, 10, 0, lane),
      aW3_1 = load_a_q(W3, 15, 10, 1, lane),
      aW3_2 = load_a_q(W3, 15, 10, 2, lane);
  v2f aW4_0 = load_a_q(W4, 10, 15, 0, lane),
      aW4_1 = load_a_q(W4, 10, 15, 1, lane),
      aW4_2 = load_a_q(W4, 10, 15, 2, lane),
      aW4_3 = load_a_q(W4, 10, 15, 3, lane);
  v2f aW5_0 = load_a(W5, 3, 10, 0, lane), aW5_1 = load_a(W5, 3, 10, 1, lane),
      aW5_2 = load_a(W5, 3, 10, 2, lane);
  v2f aV1 = load_a_q(V1, 4, 2, 0, lane);
  v2f aV2 = load_a_q(V2, 6, 4, 0, lane);
  v2f aV3_0 = load_a(V3, 4, 6, 0, lane), aV3_1 = load_a(V3, 4, 6, 1, lane);

  v8f cb1 = load_bias_q(b1, 5, lane);
  v8f cb2 = load_bias_q(b2, 10, lane);
  v8f cb3 = load_bias_q(b3, 15, lane);
  v8f cb4 = load_bias_q(b4, 10, lane);
  v8f cb5 = load_bias(b5, 3, lane);
  v8f cc1 = load_bias_q(c1, 4, lane);
  v8f cc2 = load_bias_q(c2, 6, lane);
  v8f cc3 = load_bias(c3, 4, lane);

  const float r10 = reg1[0], r11 = reg1[1], r12 = reg1[2], r13 = reg1[3],
              r14 = reg1[4], r15 = reg1[5];
  const float r20 = reg2[0], r21 = reg2[1], r22 = reg2[2], r23 = reg2[3];

  const int s = lane & 15;  // sample (row) owned by this lane
  const int rb = inBase + s * DIM;
  const int ob = outBase + s * DIM;
  const bool hi = (lane & 16) != 0;

#pragma unroll 2
  for (int j = 0; j < J18; ++j) {
    const int c0 = (2 * j + DIM - 2) % DIM;  // u[(2j-2)%36] = u1[j-1]
    const int c1c = 2 * j;                   // u[2j]        = u1[j]
    const int c2 = (2 * j + 2) % DIM;        // u[(2j+2)%36] = u1[j+1]
    float f0 = lds[rb + c0];
    float f1 = lds[rb + c1c];
    float f2 = lds[rb + c2];

    // B operand for MLP1 layer1 (K rows {0,1}/{2,3} across lane halves, pad 0)
    v2f bx;
    bx.x = hi ? f2 : f0;
    bx.y = hi ? 0.0f : f1;
    // B operand for MLP2 layer1
    v2f by;
    by.x = hi ? 0.0f : f1;
    by.y = hi ? 0.0f : f2;

    // ---- MLP1: 3 -> 5 -> 10 -> 15 -> 10 -> 3 : 13 fp32 WMMAs, no
    //      cross-lane movement, dense K chunks ----
    v8f d1 = reluN<4>(wmma4(aW1, bx, cb1));  // I_next=5 -> chunks 0,1

    v8f d2 = wmma4(aW2_0, bpair<0>(d1), cb2);
    d2 = reluN<6>(wmma4(aW2_1, bpair<1>(d1), d2));  // I_next=10 -> 3 chunks

    v8f d3 = wmma4(aW3_0, bpair<0>(d2), cb3);
    d3 = wmma4(aW3_1, bpair<1>(d2), d3);
    d3 = reluN<8>(wmma4(aW3_2, bpair<2>(d2), d3));  // I_next=15 -> 4 chunks

    v8f d4 = wmma4(aW4_0, bpair<0>(d3), cb4);
    d4 = wmma4(aW4_1, bpair<1>(d3), d4);
    d4 = wmma4(aW4_2, bpair<2>(d3), d4);
    d4 = reluN<6>(wmma4(aW4_3, bpair<3>(d3), d4));  // I_next=10 -> 3 chunks

    v8f d5 = wmma4(aW5_0, bpair<0>(d4), cb5);
    d5 = wmma4(aW5_1, bpair<1>(d4), d5);
    d5 = wmma4(aW5_2, bpair<2>(d4), d5);

    // ---- MLP2: 2 -> 4 -> 6 -> 4 : 4 WMMAs ----
    v8f e1 = reluN<2>(wmma4(aV1, by, cc1));         // I_next=4 -> 1 chunk
    v8f e2 = reluN<4>(wmma4(aV2, bpair<0>(e1), cc2));  // I_next=6 -> 2 chunks
    v8f e3 = wmma4(aV3_0, bpair<0>(e2), cc3);
    e3 = wmma4(aV3_1, bpair<1>(e2), e3);

    // ---- stencil + regression combine; lane L (<16) holds sample L's
    //      MLP outputs in d5[0..2] / e3[0..3] ----
    if (!hi) {
      float um1 = lds[rb + (2 * j + DIM - 1) % DIM];  // u[(2j-1)%36]
      float uo = lds[rb + 2 * j + 1];                 // u[2j+1]
      float up3 = lds[rb + (2 * j + 3) % DIM];        // u[(2j+3)%36]
      float um2 = f0, ue = f1, up2 = f2;
      float reg_even = r10 + r11 * ue + r12 * uo + r13 * (um2 * um1) +
                       r14 * (um1 * up2) + r15 * (ue * uo);
      float reg_odd = r20 + r21 * uo + r22 * (um1 * ue) + r23 * (ue * up2);
      float net_even = d5[0] + d5[1] * um1 + d5[2] * uo;
      float net_odd = e3[0] + e3[1] * um1 + e3[2] * uo + e3[3] * up3;
      *reinterpret_cast<float2*>(&lds[ob + 2 * j]) =
          make_float2(net_even + reg_even, net_odd + reg_odd);
    }
  }
  __syncthreads();

  // --- coalesced float4 store of the wave's 16x36 output tile ---
  float4* dst4 = reinterpret_cast<float4*>(out + elemBase);
  const float4* lout4 = reinterpret_cast<const float4*>(&lds[outBase]);
#pragma unroll
  for (int i = lane; i < TILE_V4; i += 32)
    if (i < avail4) dst4[i] = lout4[i];
}

extern "C" void kernel_launch(void* const* d_in, const int* in_sizes, int n_in,
                              void* d_out, int out_size, void* d_ws,
                              size_t ws_size, hipStream_t stream) {
  (void)n_in; (void)out_size; (void)d_ws; (void)ws_size;
  const float* u = (const float*)d_in[1];
  const float* reg1 = (const float*)d_in[2];
  const float* reg2 = (const float*)d_in[3];
  const float* W1 = (const float*)d_in[4];
  const float* b1 = (const float*)d_in[5];
  const float* W2 = (const float*)d_in[6];
  const float* b2 = (const float*)d_in[7];
  const float* W3 = (const float*)d_in[8];
  const float* b3 = (const float*)d_in[9];
  const float* W4 = (const float*)d_in[10];
  const float* b4 = (const float*)d_in[11];
  const float* W5 = (const float*)d_in[12];
  const float* b5 = (const float*)d_in[13];
  const float* V1 = (const float*)d_in[14];
  const float* c1 = (const float*)d_in[15];
  const float* V2 = (const float*)d_in[16];
  const float* c2 = (const float*)d_in[17];
  const float* V3 = (const float*)d_in[18];
  const float* c3 = (const float*)d_in[19];
  float* out = (float*)d_out;
  int N = in_sizes[1] / DIM;
  int rowsPerBlock = WAVES * ROWS_PER_WAVE;  // 64
  int grid = (N + rowsPerBlock - 1) / rowsPerBlock;
  mixmodel_kernel<<<grid, WAVES * 32, 0, stream>>>(
      u, out, reg1, reg2, W1, b1, W2, b2, W3, b3, W4, b4, W5, b5, V1, c1, V2,
      c2, V3, c3, N);
}